// InvariantPointAttention_56392920596780
// MI455X (gfx1250) — compile-verified
//
#include <hip/hip_runtime.h>
#include <hip/hip_bf16.h>
#include <math.h>

// ---------------------------------------------------------------------------
// Invariant Point Attention for MI455X (gfx1250), fp32 via V_WMMA_F32_16X16X4_F32
// b=1, n=1024, DIM=384, H=12, SK=SV=16, PK=4, PV=8, PD=128
// Roofline: pairwise_repr (512MB) must be streamed twice -> ~44us @23.3TB/s.
// Both pairwise passes stage tiles into LDS with the Tensor Data Mover
// (tensor_load_to_lds + s_wait_tensorcnt), double-buffered in the attn@pair
// pass. All matrix math uses wave32 WMMA 16x16 tiles.
// ---------------------------------------------------------------------------

typedef float v2f __attribute__((ext_vector_type(2)));
typedef float v8f __attribute__((ext_vector_type(8)));
typedef unsigned int u32x4 __attribute__((ext_vector_type(4)));
typedef int i32x4 __attribute__((ext_vector_type(4)));
typedef int i32x8 __attribute__((ext_vector_type(8)));

#define N_TOK 1024
#define HEADS 12
#define SCALAR_SCALE 0.14433756729740643f   /* (3*16)^-0.5 */
#define PAIR_SCALE   0.57735026918962576f   /* 3^-0.5      */
#define SA_LD 1028                          /* attn row stride in LDS (bank-conflict pad) */

__device__ __forceinline__ v8f wmma4(v2f a, v2f b, v8f c) {
  // (neg_a, A, neg_b, B, c_mod, C, reuse_a, reuse_b)
  return __builtin_amdgcn_wmma_f32_16x16x4_f32(false, a, false, b, (short)0, c,
                                               false, false);
}

// TDM 2-D tile load: global (row-major, fp32) -> LDS. Descriptor per ISA §8:
// group0 = {count=1 | lds_addr | global_addr[56:0] | type=2}
// group1 = {data_size=4B, pad ctl, tensor_dim0/1, tile_dim0/1, dim0 stride}
// groups 2/3 (and trailing group) zero: <=2D tile. Tracked by TENSORcnt.
__device__ __forceinline__ void tdm_load_2d(unsigned lds_addr, const void* gaddr,
                                            unsigned tensor_d0, unsigned tensor_d1,
                                            unsigned tile_d0, unsigned tile_d1,
                                            unsigned d0_stride,
                                            unsigned pad_interval_code,
                                            unsigned pad_amount_code,
                                            bool pad_en) {
  unsigned long long ga = (unsigned long long)(uintptr_t)gaddr;
  u32x4 g0;
  g0[0] = 1u;                                               // count=1, user mode
  g0[1] = lds_addr;                                         // LDS byte address
  g0[2] = (unsigned)(ga & 0xFFFFFFFFu);                     // global_addr[31:0]
  g0[3] = (unsigned)((ga >> 32) & 0x1FFFFFFu) | (2u << 30); // addr[56:32] | type=2
  i32x8 g1;
  unsigned w0 = (2u << 16);                                 // data_size = 4 bytes
  if (pad_en)
    w0 |= (1u << 20) | (pad_interval_code << 22) | (pad_amount_code << 25);
  g1[0] = (int)w0;
  g1[1] = (int)((tensor_d0 & 0xFFFFu) << 16);               // tensor_dim0 lo
  g1[2] = (int)(((tensor_d0 >> 16) & 0xFFFFu) | ((tensor_d1 & 0xFFFFu) << 16));
  g1[3] = (int)(((tensor_d1 >> 16) & 0xFFFFu) | ((tile_d0 & 0xFFFFu) << 16));
  g1[4] = (int)(tile_d1 & 0xFFFFu);                         // tile_dim1, tile_dim2=0
  g1[5] = (int)d0_stride;                                   // tensor_dim0_stride lo
  g1[6] = 0;
  g1[7] = 0;
  i32x4 z4 = {0, 0, 0, 0};
  i32x8 z8 = {0, 0, 0, 0, 0, 0, 0, 0};
  __builtin_amdgcn_tensor_load_to_lds(g0, g1, z4, z4, z8, 0);
}

// Generic row-major fp32 GEMM: C[M,N] = A[M,K] @ B[K,N] (+ bias[N]).
// One wave per 16x16 C tile. M,N multiples of 16; K multiple of 4.
__global__ void gemm_f32_wmma(const float* __restrict__ A,
                              const float* __restrict__ B,
                              const float* __restrict__ bias,
                              float* __restrict__ C, int M, int N, int K) {
  (void)M;
  const int lane = threadIdx.x & 31;
  const int n0 = blockIdx.x * 16;
  const int m0 = blockIdx.y * 16;
  const int m  = lane & 15;
  const int kk = (lane >> 4) * 2;
  v8f acc = {};
  for (int k0 = 0; k0 < K; k0 += 4) {
    v2f a, b;
    const float* ap = A + (size_t)(m0 + m) * K + (k0 + kk);
    a.x = ap[0];
    a.y = ap[1];
    const float* bp = B + (size_t)(k0 + kk) * N + n0 + m;
    b.x = bp[0];
    b.y = bp[N];
    acc = wmma4(a, b, acc);
  }
  const float bv = bias ? bias[n0 + m] : 0.0f;
  const int hibase = (lane >> 4) * 8;
#pragma unroll
  for (int r = 0; r < 8; ++r) {
    const int mm = r + hibase;
    C[(size_t)(m0 + mm) * N + n0 + m] = acc[r] + bv;
  }
}

// Build per-head packed K=32 fragments for the logits GEMM, plus per-head
// value layouts.
//   Acat[h][i][0..15]=q_s*SCALAR_SCALE  [16..27]=q_p*pw  [28]=-0.5*pw*q2 [29]=1
//   Bcat[h][j][0..15]=k_s               [16..27]=k_p     [28]=1
//                     [29]=-0.5*pw*k2 + PAIR_SCALE*b_pb[h]
__global__ void build_frags(const float* __restrict__ qs,
                            const float* __restrict__ ks,
                            const float* __restrict__ vs,
                            const float* __restrict__ qp,
                            const float* __restrict__ kp,
                            const float* __restrict__ vp,
                            const float* __restrict__ point_w,
                            const float* __restrict__ b_pb,
                            float* __restrict__ Acat, float* __restrict__ Bcat,
                            float* __restrict__ vsh, float* __restrict__ vph) {
  const int idx = blockIdx.x * blockDim.x + threadIdx.x;
  if (idx >= HEADS * N_TOK) return;
  const int h = idx >> 10;
  const int i = idx & (N_TOK - 1);
  const float pw = log1pf(expf(point_w[h]));  // softplus

  float* Ar = Acat + ((size_t)h * N_TOK + i) * 32;
  float* Br = Bcat + ((size_t)h * N_TOK + i) * 32;

#pragma unroll
  for (int t = 0; t < 16; ++t) {
    Ar[t] = qs[(size_t)i * 192 + h * 16 + t] * SCALAR_SCALE;
    Br[t] = ks[(size_t)i * 192 + h * 16 + t];
  }
  float q2 = 0.0f, k2 = 0.0f;
#pragma unroll
  for (int t = 0; t < 12; ++t) {
    const float qv = qp[(size_t)i * 144 + h * 12 + t];
    const float kv = kp[(size_t)i * 144 + h * 12 + t];
    q2 += qv * qv;
    k2 += kv * kv;
    Ar[16 + t] = qv * pw;
    Br[16 + t] = kv;
  }
  Ar[28] = -0.5f * pw * q2;
  Br[28] = 1.0f;
  Ar[29] = 1.0f;
  Br[29] = -0.5f * pw * k2 + b_pb[h] * PAIR_SCALE;
  Ar[30] = 0.0f; Ar[31] = 0.0f;
  Br[30] = 0.0f; Br[31] = 0.0f;

#pragma unroll
  for (int t = 0; t < 16; ++t)
    vsh[((size_t)h * N_TOK + i) * 16 + t] = vs[(size_t)i * 192 + h * 16 + t];
#pragma unroll
  for (int t = 0; t < 24; ++t)
    vph[((size_t)h * N_TOK + i) * 24 + t] = vp[(size_t)i * 288 + h * 24 + t];
}

// logits[h][i][j] = Acat[h][i][:32] . Bcat[h][j][:32]   (one wave / 16x16 tile)
__global__ void logits_gemm(const float* __restrict__ Acat,
                            const float* __restrict__ Bcat,
                            float* __restrict__ logits) {
  const int lane = threadIdx.x & 31;
  const int j0 = blockIdx.x * 16;
  const int i0 = blockIdx.y * 16;
  const int h  = blockIdx.z;
  const float* A = Acat + (size_t)h * N_TOK * 32;
  const float* B = Bcat + (size_t)h * N_TOK * 32;
  const int m  = lane & 15;
  const int kk = (lane >> 4) * 2;
  v8f acc = {};
#pragma unroll
  for (int k0 = 0; k0 < 32; k0 += 4) {
    v2f a, b;
    const float* ap = A + (size_t)(i0 + m) * 32 + (k0 + kk);
    a.x = ap[0];
    a.y = ap[1];
    const float* bp = B + (size_t)(j0 + m) * 32 + (k0 + kk);  // B^T access
    b.x = bp[0];
    b.y = bp[1];
    acc = wmma4(a, b, acc);
  }
  float* L = logits + (size_t)h * N_TOK * N_TOK;
  const int hibase = (lane >> 4) * 8;
#pragma unroll
  for (int r = 0; r < 8; ++r) {
    const int mm = r + hibase;
    L[(size_t)(i0 + mm) * N_TOK + j0 + m] = acc[r];
  }
}

// logits[h][i][j] += PAIR_SCALE * (pairwise[i][j][:] . W_pb[:,h])
// Block = 8 waves: TDM-stage pairwise[i][jb*128 .. +128][0..127] into LDS with
// hardware row padding (130-float stride -> conflict-free column reads), then
// each wave does a 16x16x128 WMMA slice (M=heads, N=j, K=pd).
__global__ void bias_add(const float* __restrict__ pair,
                         const float* __restrict__ Wpb,
                         float* __restrict__ logits) {
  extern __shared__ float sP[];   // 128 rows x 130 floats (TDM pad 2 dw/row)
  const int jb = blockIdx.x;      // 128-wide j block
  const int i  = blockIdx.y;
  const int t  = threadIdx.x;
  if (t < 32) {
    tdm_load_2d((unsigned)(uintptr_t)sP,
                pair + ((size_t)i * N_TOK + (size_t)jb * 128) * 128,
                /*tensor_d0=*/128, /*tensor_d1=*/N_TOK,
                /*tile_d0=*/128, /*tile_d1=*/128, /*d0_stride=*/128,
                /*pad_interval: 128 dwords*/ 6, /*pad_amount: 2 dwords*/ 1, true);
    __builtin_amdgcn_s_wait_tensorcnt(0);
  }
  __syncthreads();
  const int wave = t >> 5;           // j sub-tile 0..7
  const int lane = t & 31;
  const int m  = lane & 15;          // h for A rows, j for B cols
  const int kk = (lane >> 4) * 2;
  const int hm = (m < HEADS) ? m : 0;
  const float asc = (m < HEADS) ? PAIR_SCALE : 0.0f;
  const float* brow = sP + (wave * 16 + m) * 130;
  v8f acc = {};
#pragma unroll 4
  for (int k0 = 0; k0 < 128; k0 += 4) {
    v2f a, b;
    a.x = Wpb[(size_t)(k0 + kk) * HEADS + hm] * asc;        // W_pb^T row h
    a.y = Wpb[(size_t)(k0 + kk + 1) * HEADS + hm] * asc;
    b.x = brow[k0 + kk];
    b.y = brow[k0 + kk + 1];
    acc = wmma4(a, b, acc);
  }
  const int j = jb * 128 + wave * 16 + m;
  const int hibase = (lane >> 4) * 8;
#pragma unroll
  for (int r = 0; r < 8; ++r) {
    const int h = r + hibase;
    if (h < HEADS) {
      logits[((size_t)h * N_TOK + i) * N_TOK + j] += acc[r];
    }
  }
}

// Row softmax over j; one 256-thread block per (h,i) row of 1024.
__global__ void softmax_rows(float* __restrict__ logits) {
  __shared__ float sred[8];
  const int t = threadIdx.x;
  const int wid = t >> 5, lane = t & 31;
  float4* p = (float4*)(logits + (size_t)blockIdx.x * N_TOK);
  float4 v = p[t];
  float mx = fmaxf(fmaxf(v.x, v.y), fmaxf(v.z, v.w));
#pragma unroll
  for (int o = 16; o; o >>= 1) mx = fmaxf(mx, __shfl_xor(mx, o, 32));
  if (lane == 0) sred[wid] = mx;
  __syncthreads();
  float bm = sred[0];
#pragma unroll
  for (int w = 1; w < 8; ++w) bm = fmaxf(bm, sred[w]);
  v.x = __expf(v.x - bm);
  v.y = __expf(v.y - bm);
  v.z = __expf(v.z - bm);
  v.w = __expf(v.w - bm);
  float s = v.x + v.y + v.z + v.w;
#pragma unroll
  for (int o = 16; o; o >>= 1) s += __shfl_xor(s, o, 32);
  __syncthreads();
  if (lane == 0) sred[wid] = s;
  __syncthreads();
  float bs = 0.0f;
#pragma unroll
  for (int w = 0; w < 8; ++w) bs += sred[w];
  const float inv = 1.0f / bs;
  v.x *= inv; v.y *= inv; v.z *= inv; v.w *= inv;
  p[t] = v;
}

// res_scalar + res_point: per head, attn[h](1024x1024) @ V, writing directly
// into the concatenated `results` layout. blockIdx.x: 0 -> v_s cols 0..15,
// 1 -> v_p cols 0..15, 2 -> v_p cols 16..23.
__global__ void attn_v_gemm(const float* __restrict__ attn,
                            const float* __restrict__ vsh,
                            const float* __restrict__ vph,
                            float* __restrict__ results) {
  const int lane = threadIdx.x & 31;
  const int tile = blockIdx.x;
  const int i0 = blockIdx.y * 16;
  const int h  = blockIdx.z;
  const int m  = lane & 15;
  const int kk = (lane >> 4) * 2;
  const float* A = attn + (size_t)h * N_TOK * N_TOK;
  const float* Bsrc;
  int ldb, ncol0, nvalid;
  if (tile == 0) { Bsrc = vsh + (size_t)h * N_TOK * 16; ldb = 16; ncol0 = 0;  nvalid = 16; }
  else if (tile == 1) { Bsrc = vph + (size_t)h * N_TOK * 24; ldb = 24; ncol0 = 0;  nvalid = 16; }
  else { Bsrc = vph + (size_t)h * N_TOK * 24; ldb = 24; ncol0 = 16; nvalid = 8; }
  const int nc = (m < nvalid) ? m : 0;
  v8f acc = {};
  for (int k0 = 0; k0 < N_TOK; k0 += 4) {
    v2f a, b;
    const float* ap = A + (size_t)(i0 + m) * N_TOK + (k0 + kk);
    a.x = ap[0];
    a.y = ap[1];
    const float* bp = Bsrc + (size_t)(k0 + kk) * ldb + ncol0 + nc;
    const float bx = bp[0], by = bp[ldb];
    b.x = (m < nvalid) ? bx : 0.0f;
    b.y = (m < nvalid) ? by : 0.0f;
    acc = wmma4(a, b, acc);
  }
  const int hibase = (lane >> 4) * 8;
#pragma unroll
  for (int r = 0; r < 8; ++r) {
    const int mm = r + hibase;
    if (m < nvalid) {
      size_t col;
      if (tile == 0) col = (size_t)h * 16 + m;                       // scalar block
      else           col = 192 + 1536 + (size_t)h * 24 + ncol0 + m;  // point block
      results[(size_t)(i0 + mm) * 2016 + col] = acc[r];
    }
  }
}

// res_pair: per row i, stage attn[0..11][i][:] in LDS (padded rows, no bank
// conflicts) and double-buffer 64x128 pairwise k-tiles via TDM; 8 waves each
// compute a 16-wide column slice of attn(12x1024) @ pairwise[i](1024x128).
__global__ void res_pair_gemm(const float* __restrict__ attn,
                              const float* __restrict__ pair,
                              float* __restrict__ results) {
  extern __shared__ float smem[];
  float* sA = smem;                          // HEADS x SA_LD
  float* sB = smem + HEADS * SA_LD;          // 2 x (64 x 128)
  const int i = blockIdx.x;
  const int t = threadIdx.x;  // 256 threads = 8 waves
  const float* prow = pair + (size_t)i * N_TOK * 128;
  // issue first TDM tile, then stage attn rows while it flies
  if (t < 32) {
    tdm_load_2d((unsigned)(uintptr_t)sB, prow,
                128, N_TOK, 128, 64, 128, 0, 0, false);
  }
  for (int h = 0; h < HEADS; ++h) {
    const float4 v =
        ((const float4*)(attn + ((size_t)h * N_TOK + i) * N_TOK))[t];
    ((float4*)(sA + h * SA_LD))[t] = v;
  }
  const int wave = t >> 5;  // column tile 0..7 (n0 = wave*16)
  const int lane = t & 31;
  const int m  = lane & 15;
  const int kk = (lane >> 4) * 2;
  const int n0 = wave * 16;
  const int hm = (m < HEADS) ? m : 0;
  v8f acc = {};
  for (int tt = 0; tt < 16; ++tt) {
    if (t < 32) {
      if (tt + 1 < 16) {
        tdm_load_2d((unsigned)(uintptr_t)(sB + ((tt + 1) & 1) * 64 * 128),
                    prow + (size_t)(tt + 1) * 64 * 128,
                    128, N_TOK, 128, 64, 128, 0, 0, false);
        __builtin_amdgcn_s_wait_tensorcnt(1);
      } else {
        __builtin_amdgcn_s_wait_tensorcnt(0);
      }
    }
    __syncthreads();  // tile tt visible to all waves (and sA on tt==0)
    const float* bufc = sB + (tt & 1) * 64 * 128;
    const int kg = tt * 64;
#pragma unroll 4
    for (int kl = 0; kl < 64; kl += 4) {
      v2f a, b;
      const float ax = sA[hm * SA_LD + kg + kl + kk];
      const float ay = sA[hm * SA_LD + kg + kl + kk + 1];
      a.x = (m < HEADS) ? ax : 0.0f;
      a.y = (m < HEADS) ? ay : 0.0f;
      b.x = bufc[(kl + kk) * 128 + n0 + m];
      b.y = bufc[(kl + kk + 1) * 128 + n0 + m];
      acc = wmma4(a, b, acc);
    }
    __syncthreads();  // all waves done with buffer before it is re-filled
  }
  const int hibase = (lane >> 4) * 8;
#pragma unroll
  for (int r = 0; r < 8; ++r) {
    const int h = r + hibase;
    if (h < HEADS)
      results[(size_t)i * 2016 + 192 + (size_t)h * 128 + n0 + m] = acc[r];
  }
}

extern "C" void kernel_launch(void* const* d_in, const int* in_sizes, int n_in,
                              void* d_out, int out_size, void* d_ws,
                              size_t ws_size, hipStream_t stream) {
  (void)in_sizes; (void)n_in; (void)out_size; (void)ws_size;
  const float* x     = (const float*)d_in[0];
  const float* pairw = (const float*)d_in[1];
  // d_in[2] rotations, d_in[3] translations: unused by reference
  const float* W_sq  = (const float*)d_in[4];
  const float* W_sk  = (const float*)d_in[5];
  const float* W_sv  = (const float*)d_in[6];
  const float* W_pq  = (const float*)d_in[7];
  const float* W_pk  = (const float*)d_in[8];
  const float* W_pv  = (const float*)d_in[9];
  const float* ptw   = (const float*)d_in[10];
  const float* W_pb  = (const float*)d_in[11];
  const float* b_pb  = (const float*)d_in[12];
  const float* W_out = (const float*)d_in[13];
  const float* b_out = (const float*)d_in[14];
  float* out = (float*)d_out;

  // workspace carve-up (floats); total ~17.1M floats (~68.5 MB)
  float* ws = (float*)d_ws;
  size_t off = 0;
  float* qs      = ws + off; off += (size_t)N_TOK * 192;
  float* ks_     = ws + off; off += (size_t)N_TOK * 192;
  float* vs      = ws + off; off += (size_t)N_TOK * 192;
  float* qp      = ws + off; off += (size_t)N_TOK * 144;
  float* kp      = ws + off; off += (size_t)N_TOK * 144;
  float* vp      = ws + off; off += (size_t)N_TOK * 288;
  float* Acat    = ws + off; off += (size_t)HEADS * N_TOK * 32;
  float* Bcat    = ws + off; off += (size_t)HEADS * N_TOK * 32;
  float* vsh     = ws + off; off += (size_t)HEADS * N_TOK * 16;
  float* vph     = ws + off; off += (size_t)HEADS * N_TOK * 24;
  float* logits  = ws + off; off += (size_t)HEADS * N_TOK * N_TOK;
  float* results = ws + off; off += (size_t)N_TOK * 2016;

  const dim3 wave(32);
  // projections: x(1024x384) @ W(384xNo)
  gemm_f32_wmma<<<dim3(12, 64), wave, 0, stream>>>(x, W_sq, nullptr, qs,  N_TOK, 192, 384);
  gemm_f32_wmma<<<dim3(12, 64), wave, 0, stream>>>(x, W_sk, nullptr, ks_, N_TOK, 192, 384);
  gemm_f32_wmma<<<dim3(12, 64), wave, 0, stream>>>(x, W_sv, nullptr, vs,  N_TOK, 192, 384);
  gemm_f32_wmma<<<dim3(9,  64), wave, 0, stream>>>(x, W_pq, nullptr, qp,  N_TOK, 144, 384);
  gemm_f32_wmma<<<dim3(9,  64), wave, 0, stream>>>(x, W_pk, nullptr, kp,  N_TOK, 144, 384);
  gemm_f32_wmma<<<dim3(18, 64), wave, 0, stream>>>(x, W_pv, nullptr, vp,  N_TOK, 288, 384);

  build_frags<<<(HEADS * N_TOK + 255) / 256, 256, 0, stream>>>(
      qs, ks_, vs, qp, kp, vp, ptw, b_pb, Acat, Bcat, vsh, vph);

  logits_gemm<<<dim3(64, 64, HEADS), wave, 0, stream>>>(Acat, Bcat, logits);

  const size_t shmem_bias = 128 * 130 * sizeof(float);  // 65 KB (TDM-padded)
  bias_add<<<dim3(8, N_TOK), 256, shmem_bias, stream>>>(pairw, W_pb, logits);

  softmax_rows<<<HEADS * N_TOK, 256, 0, stream>>>(logits);

  attn_v_gemm<<<dim3(3, 64, HEADS), wave, 0, stream>>>(logits, vsh, vph, results);

  const size_t shmem_rp = (HEADS * SA_LD + 2 * 64 * 128) * sizeof(float);
  res_pair_gemm<<<N_TOK, 256, shmem_rp, stream>>>(logits, pairw, results);

  // final: results(1024x2016) @ W_out(2016x384) + b_out
  gemm_f32_wmma<<<dim3(24, 64), wave, 0, stream>>>(results, W_out, b_out, out,
                                                   N_TOK, 384, 2016);
}